// VectorQuantizer_55903294324894
// MI455X (gfx1250) — compile-verified
//
#include <hip/hip_runtime.h>

typedef float v2f __attribute__((ext_vector_type(2)));
typedef float v8f __attribute__((ext_vector_type(8)));

#define D            64
#define K            1024
#define ESTRIDE      66          // padded row stride (floats): keeps float2 LDS reads 8B-aligned, spreads banks
#define ROWS_TOTAL   65536
#define THREADS      256
#define WAVES        8
#define SLABS        2           // 16-row slabs per wave
#define ROWS_PER_BLOCK (WAVES * 16 * SLABS)          // 256
#define NUM_BLOCKS   (ROWS_TOTAL / ROWS_PER_BLOCK)   // 256

#define LDS_E_WORDS  (K * ESTRIDE)                   // 67584
#define LDS_ESQ_OFF  LDS_E_WORDS
#define LDS_WL_OFF   (LDS_E_WORDS + K)
#define LDS_WORDS    (LDS_E_WORDS + K + WAVES)       // ~270 KB < 320 KB WGP LDS

__global__ __launch_bounds__(THREADS)
void vq_main(const float* __restrict__ x, const float* __restrict__ emb,
             float* __restrict__ out, float* __restrict__ partial)
{
    extern __shared__ float lds[];
    float* eLds = lds;                 // [K][ESTRIDE] codebook
    float* esq  = lds + LDS_ESQ_OFF;   // [K] ||e_k||^2
    float* wl   = lds + LDS_WL_OFF;    // [WAVES] loss partials

    const int tid  = threadIdx.x;
    const int wave = tid >> 5;
    const int lane = tid & 31;
    const int n    = lane & 15;        // column/row stripe within half-wave
    const int hi   = lane >> 4;        // half-wave select

    // ---- stage full codebook into LDS (coalesced global, conflict-free LDS writes)
    for (int i = tid; i < K * D; i += THREADS)
        eLds[(i >> 6) * ESTRIDE + (i & 63)] = emb[i];
    __syncthreads();

    // ---- per-code squared norms, wave-cooperative (conflict-free LDS reads)
    for (int c = wave; c < K; c += WAVES) {
        const float* er = eLds + c * ESTRIDE;
        float v0 = er[lane], v1 = er[lane + 32];
        float s = v0 * v0 + v1 * v1;
        #pragma unroll
        for (int m = 16; m >= 1; m >>= 1) s += __shfl_xor(s, m, 32);
        if (lane == 0) esq[c] = s;
    }
    __syncthreads();

    float waveLoss = 0.0f;

    for (int slab = 0; slab < SLABS; ++slab) {
        const int rowBase = blockIdx.x * ROWS_PER_BLOCK + slab * (WAVES * 16) + wave * 16;

        // ---- A fragments (16x4 f32 layout): lane (m=n, hi) holds K = {4k+2hi, 4k+2hi+1}
        const float* xrow = x + (size_t)(rowBase + n) * D + 2 * hi;
        v2f a[16];
        float xsqp = 0.0f;
        #pragma unroll
        for (int kk = 0; kk < 16; ++kk) {
            a[kk] = *(const v2f*)(xrow + 4 * kk);
            xsqp += a[kk][0] * a[kk][0] + a[kk][1] * a[kk][1];
        }
        const float xsqFull = xsqp + __shfl_xor(xsqp, 16, 32);  // full ||x_row||^2, row = rowBase+n

        float best[8];
        int   bidx[8];
        #pragma unroll
        for (int j = 0; j < 8; ++j) { best[j] = __builtin_inff(); bidx[j] = 0; }

        // ---- sweep all 1024 codes in 16-wide tiles; distances via fp32 WMMA
        for (int t = 0; t < K / 16; ++t) {
            const int c0 = t * 16;
            const float* brow = eLds + (c0 + n) * ESTRIDE + 2 * hi;  // B (4x16): E^T fragment
            v8f acc = {0.f, 0.f, 0.f, 0.f, 0.f, 0.f, 0.f, 0.f};
            #pragma unroll
            for (int kk = 0; kk < 16; ++kk) {
                v2f b = *(const v2f*)(brow + 4 * kk);
                acc = __builtin_amdgcn_wmma_f32_16x16x4_f32(
                          false, a[kk], false, b, (short)0, acc, false, false);
            }
            // C layout: VGPR j -> row (j + 8*hi), col = c0 + n
            const float es  = esq[c0 + n];
            const int   code = c0 + n;
            #pragma unroll
            for (int j = 0; j < 8; ++j) {
                float sc = __builtin_fmaf(-2.0f, acc[j], es);  // ||e||^2 - 2 x.e
                if (sc < best[j]) { best[j] = sc; bidx[j] = code; }
            }
        }

        // ---- min+argmin butterfly across the 16 lanes of each half-wave (first-min tiebreak)
        #pragma unroll
        for (int j = 0; j < 8; ++j) {
            #pragma unroll
            for (int m = 1; m <= 8; m <<= 1) {
                float ob = __shfl_xor(best[j], m, 32);
                int   oi = __shfl_xor(bidx[j], m, 32);
                if (ob < best[j] || (ob == best[j] && oi < bidx[j])) {
                    best[j] = ob; bidx[j] = oi;
                }
            }
        }
        // lane (n,hi): best/bidx[j] now hold result for row (rowBase + j + 8*hi)

        // ---- slab loss = sum_rows(min_score) + sum_rows(||x||^2)
        float lp = (hi == 0) ? xsqFull : 0.0f;     // rows' xsq live on lanes 0..15
        if (n == 0) {
            float s8 = 0.0f;
            #pragma unroll
            for (int j = 0; j < 8; ++j) s8 += best[j];
            lp += s8;                               // lanes 0 and 16 carry 8 rows each
        }
        #pragma unroll
        for (int m = 16; m >= 1; m >>= 1) lp += __shfl_xor(lp, m, 32);
        waveLoss += lp;

        // ---- write quantized rows: out[row] = E[argmin]
        #pragma unroll
        for (int r = 0; r < 16; ++r) {
            const int j = r & 7;
            const int srcLane = (r >> 3) << 4;                 // lane 0 or 16
            const int idx = __shfl(bidx[j], srcLane, 32);
            const float* er = eLds + idx * ESTRIDE;
            float* orow = out + (size_t)(rowBase + r) * D;
            orow[lane]      = er[lane];
            orow[lane + 32] = er[lane + 32];
        }
    }

    if (lane == 0) wl[wave] = waveLoss;
    __syncthreads();
    if (tid == 0) {
        float s = 0.0f;
        #pragma unroll
        for (int w = 0; w < WAVES; ++w) s += wl[w];
        partial[blockIdx.x] = s;
    }
}

__global__ __launch_bounds__(NUM_BLOCKS)
void vq_finalize(const float* __restrict__ partial, float* __restrict__ lossOut)
{
    __shared__ float red[NUM_BLOCKS];
    const int t = threadIdx.x;
    red[t] = partial[t];
    __syncthreads();
    for (int s = NUM_BLOCKS / 2; s > 0; s >>= 1) {
        if (t < s) red[t] += red[t + s];
        __syncthreads();
    }
    if (t == 0)
        lossOut[0] = 1.25f * red[0] / (float)((size_t)ROWS_TOTAL * D);
}

extern "C" void kernel_launch(void* const* d_in, const int* in_sizes, int n_in,
                              void* d_out, int out_size, void* d_ws, size_t ws_size,
                              hipStream_t stream)
{
    (void)in_sizes; (void)n_in; (void)out_size; (void)ws_size;
    const float* x   = (const float*)d_in[0];
    const float* emb = (const float*)d_in[1];
    float* out       = (float*)d_out;
    float* partial   = (float*)d_ws;   // NUM_BLOCKS floats; every slot written each call

    vq_main<<<NUM_BLOCKS, THREADS, LDS_WORDS * sizeof(float), stream>>>(x, emb, out, partial);
    vq_finalize<<<1, NUM_BLOCKS, 0, stream>>>(partial, out + (size_t)ROWS_TOTAL * D);
}